// MotifPooling_68624987455945
// MI455X (gfx1250) — compile-verified
//
#include <hip/hip_runtime.h>
#include <hip/hip_bf16.h>

// ---------------------------------------------------------------------------
// MotifPooling for MI455X (gfx1250, wave32).
//   s_motif  = scatter_mean(s, seg)            [M, 256]
//   v_motif  = scatter_mean(v, seg)            [M, 16, 3]
//   s_out    = s_motif @ Ws^T + bs             [M, 256]   <- V_WMMA_F32_16X16X4_F32
//   v_out    = einsum('mcd,oc->mod') + bv      [M, 16, 3] <- VALU (tiny)
// ---------------------------------------------------------------------------

#define N_NODES 200000
#define CH      256
#define VCH     16
#define VDIM    3
#define M_SEG   50000

typedef float v2f __attribute__((ext_vector_type(2)));
typedef float v8f __attribute__((ext_vector_type(8)));

// ---------------------------------------------------------------- zero scratch
__global__ __launch_bounds__(256) void mp_zero(float* __restrict__ ws, long n) {
    long i = (long)blockIdx.x * blockDim.x + threadIdx.x;
    if (i < n) ws[i] = 0.0f;
}

// ------------------------------------------------------------- scatter (sums)
// one block (256 threads) per node; f32 atomics land in the 192MB L2.
__global__ __launch_bounds__(256) void mp_scatter(
        const float* __restrict__ s, const float* __restrict__ v,
        const int* __restrict__ seg,
        float* __restrict__ s_sum, float* __restrict__ v_sum,
        float* __restrict__ cnt) {
    int n = blockIdx.x;
    int t = threadIdx.x;
    int m = seg[n];
    unsafeAtomicAdd(&s_sum[(size_t)m * CH + t], s[(size_t)n * CH + t]);
    if (t < VCH * VDIM)
        unsafeAtomicAdd(&v_sum[(size_t)m * (VCH * VDIM) + t],
                        v[(size_t)n * (VCH * VDIM) + t]);
    if (t == 0) unsafeAtomicAdd(&cnt[m], 1.0f);
}

// --------------------------------------------------- s GEMM via f32 WMMA 16x16x4
// Each wave: one 16-row x 64-col strip of s_out (4 accumulator tiles).
// 4 waves per block cover all 256 columns; grid.x = M/16 = 3125.
//
// A-matrix (16x4 f32) VGPR layout (ISA 7.12.2):
//   lanes 0-15 : M=lane,     v0=K0, v1=K1
//   lanes 16-31: M=lane-16,  v0=K2, v1=K3
// B-matrix (4x16) mirrors it with N across lanes; B[k][n] = Ws[n][k], so each
// lane reads a contiguous float2 from row n of Ws.
__global__ __launch_bounds__(128) void mp_s_gemm(
        const float* __restrict__ s_sum, const float* __restrict__ cnt,
        const float* __restrict__ Ws, const float* __restrict__ bs,
        float* __restrict__ s_out) {
    const int wave = threadIdx.x >> 5;      // 0..3 -> column strip
    const int lane = threadIdx.x & 31;
    const int half = lane >> 4;             // 0 or 1 (K sub-pair)
    const int l15  = lane & 15;
    const int m0   = blockIdx.x * 16;
    const int col0 = wave * 64;

    const int arow = m0 + l15;
    const float inv = 1.0f / fmaxf(cnt[arow], 1.0f);   // fold scatter-mean div

    v8f acc[4] = {};
    const float* __restrict__ Aptr = s_sum + (size_t)arow * CH + 2 * half;

    for (int k0 = 0; k0 < CH; k0 += 4) {
        v2f a = *(const v2f*)(Aptr + k0);
        a.x *= inv;
        a.y *= inv;
#pragma unroll
        for (int t = 0; t < 4; ++t) {
            const int bcol = col0 + t * 16 + l15;
            v2f b = *(const v2f*)(Ws + (size_t)bcol * CH + k0 + 2 * half);
            acc[t] = __builtin_amdgcn_wmma_f32_16x16x4_f32(
                /*neg_a=*/false, a, /*neg_b=*/false, b,
                /*c_mod=*/(short)0, acc[t],
                /*reuse_a=*/false, /*reuse_b=*/false);
        }
    }

    // C/D layout: VGPR i -> row m0+i (lanes 0-15) / m0+8+i (lanes 16-31), N=l15
#pragma unroll
    for (int t = 0; t < 4; ++t) {
        const int col  = col0 + t * 16 + l15;
        const float bb = bs[col];
#pragma unroll
        for (int i = 0; i < 8; ++i) {
            const int row = m0 + i + 8 * half;
            s_out[(size_t)row * CH + col] = acc[t][i] + bb;
        }
    }
}

// ---------------------------------------------------------------- v einsum
__global__ __launch_bounds__(256) void mp_v_pool(
        const float* __restrict__ v_sum, const float* __restrict__ cnt,
        const float* __restrict__ Wv, const float* __restrict__ bv,
        float* __restrict__ v_out) {
    int i = blockIdx.x * blockDim.x + threadIdx.x;
    if (i >= M_SEG * VCH * VDIM) return;
    const int d = i % VDIM;
    const int o = (i / VDIM) % VCH;
    const int m = i / (VDIM * VCH);
    const float inv = 1.0f / fmaxf(cnt[m], 1.0f);
    const float* __restrict__ vm = v_sum + (size_t)m * (VCH * VDIM) + d;
    const float* __restrict__ w  = Wv + o * VCH;
    float acc = 0.0f;
#pragma unroll
    for (int c = 0; c < VCH; ++c) acc += w[c] * vm[c * VDIM];
    v_out[i] = acc * inv + bv[o];
}

// ---------------------------------------------------------------------------
extern "C" void kernel_launch(void* const* d_in, const int* in_sizes, int n_in,
                              void* d_out, int out_size, void* d_ws, size_t ws_size,
                              hipStream_t stream) {
    const float* s   = (const float*)d_in[0];   // [N, 256]
    const float* v   = (const float*)d_in[1];   // [N, 16, 3]
    const int*   seg = (const int*)  d_in[2];   // [N]
    const float* Ws  = (const float*)d_in[3];   // [256, 256]
    const float* bs  = (const float*)d_in[4];   // [256]
    const float* Wv  = (const float*)d_in[5];   // [16, 16]
    const float* bv  = (const float*)d_in[6];   // [16]

    float* out   = (float*)d_out;
    float* s_out = out;                               // M*256
    float* v_out = out + (size_t)M_SEG * CH;          // M*48

    // workspace: s_sum | v_sum | cnt  (~61 MB)
    float* s_sum = (float*)d_ws;
    float* v_sum = s_sum + (size_t)M_SEG * CH;
    float* cnt   = v_sum + (size_t)M_SEG * VCH * VDIM;

    const long ztot = (long)M_SEG * (CH + VCH * VDIM + 1);
    mp_zero<<<(unsigned)((ztot + 255) / 256), 256, 0, stream>>>((float*)d_ws, ztot);

    mp_scatter<<<N_NODES, 256, 0, stream>>>(s, v, seg, s_sum, v_sum, cnt);

    mp_s_gemm<<<M_SEG / 16, 128, 0, stream>>>(s_sum, cnt, Ws, bs, s_out);

    const int vtot = M_SEG * VCH * VDIM;
    mp_v_pool<<<(vtot + 255) / 256, 256, 0, stream>>>(v_sum, cnt, Wv, bv, v_out);
}